// BaseFsl_90391881712070
// MI455X (gfx1250) — compile-verified
//
#include <hip/hip_runtime.h>

typedef __attribute__((ext_vector_type(2))) float v2f;
typedef __attribute__((ext_vector_type(8))) float v8f;

#define NCLS 64
#define DIM  128
#define WS_FLOATS (NCLS * DIM + NCLS) // sums(8192) + counts(64)

// ---------------------------------------------------------------------------
// Stage 0: zero workspace (sums + counts)
// ---------------------------------------------------------------------------
__global__ void fsl_zero_ws(float* __restrict__ ws, int n) {
    int i = blockIdx.x * blockDim.x + threadIdx.x;
    if (i < n) ws[i] = 0.0f;
}

// ---------------------------------------------------------------------------
// Stage 1: per-block LDS accumulation of class sums/counts, then one global
// atomic flush per block (keeps global atomic traffic at ceil(Ns/256) blocks x 8256).
// ---------------------------------------------------------------------------
__global__ void __launch_bounds__(256)
fsl_proto_accum(const float* __restrict__ emb, const int* __restrict__ labels,
                float* __restrict__ sums, float* __restrict__ counts, int Ns) {
    __shared__ float ls[NCLS * DIM];
    __shared__ float lc[NCLS];
    int tid = threadIdx.x;
    for (int i = tid; i < NCLS * DIM; i += 256) ls[i] = 0.0f;
    if (tid < NCLS) lc[tid] = 0.0f;
    __syncthreads();

    int dim  = tid & (DIM - 1);
    int sub  = tid >> 7; // 0 or 1: two rows processed per iteration
    int base = blockIdx.x * 256;
    for (int r = base + sub; r < base + 256; r += 2) {
        if (r < Ns) {
            int lab = labels[r] & (NCLS - 1);
            atomicAdd(&ls[lab * DIM + dim], emb[(size_t)r * DIM + dim]);
            if (dim == 0) atomicAdd(&lc[lab], 1.0f);
        }
    }
    __syncthreads();
    for (int i = tid; i < NCLS * DIM; i += 256) atomicAdd(&sums[i], ls[i]);
    if (tid < NCLS) atomicAdd(&counts[tid], lc[tid]);
}

// ---------------------------------------------------------------------------
// Stage 2: finalize prototypes in place: sums -> prototypes (divide by count),
// counts -> ||prototype||^2.
// ---------------------------------------------------------------------------
__global__ void __launch_bounds__(128)
fsl_proto_final(float* __restrict__ sums, float* __restrict__ counts) {
    int c = blockIdx.x;
    int d = threadIdx.x;
    float cnt = fmaxf(counts[c], 1.0f);
    float p   = sums[c * DIM + d] / cnt;

    float pp = p * p;
    pp += __shfl_xor(pp, 16, 32);
    pp += __shfl_xor(pp, 8, 32);
    pp += __shfl_xor(pp, 4, 32);
    pp += __shfl_xor(pp, 2, 32);
    pp += __shfl_xor(pp, 1, 32);

    __shared__ float part[4];
    int lane = d & 31, w = d >> 5;
    if (lane == 0) part[w] = pp;
    __syncthreads();

    sums[c * DIM + d] = p; // sums buffer now holds prototypes
    if (d == 0) counts[c] = part[0] + part[1] + part[2] + part[3]; // -> pnorm2
}

// ---------------------------------------------------------------------------
// Stage 3: main kernel. Block = 256 threads = 8 wave32; each wave computes a
// 16(queries) x 64(classes) tile via V_WMMA_F32_16X16X4_F32 (K=128 in steps
// of 4), then does sqrt-distance + row softmax entirely in registers using
// the WMMA C-matrix layout (rows r / r+8 split across half-waves).
//
// Prototype staging (32 KB per block) uses the CDNA5 async global->LDS path
// (GLOBAL_LOAD_ASYNC_TO_LDS_B128, ASYNCcnt) instead of a VGPR round-trip.
//
// f32 WMMA operand layout (wave32):
//   A 16x4:  lanes 0-15 -> M=lane, K={0,1}; lanes 16-31 -> M=lane-16, K={2,3}
//   B 4x16:  lanes 0-15 -> N=lane, K={0,1}; lanes 16-31 -> N=lane-16, K={2,3}
//   C 16x16: VGPR r -> row r (lanes 0-15, N=lane) / row r+8 (lanes 16-31)
// ---------------------------------------------------------------------------
__global__ void __launch_bounds__(256)
fsl_query_softmax(const float* __restrict__ Q, const float* __restrict__ protos,
                  const float* __restrict__ pnorm2, float* __restrict__ out,
                  int Nq) {
    __shared__ float ldsP[NCLS * DIM]; // 32 KB prototype stage (of 320 KB WGP LDS)
    __shared__ float ldsPn[NCLS];

    int tid = threadIdx.x;

    // Async-copy prototypes to LDS: 2048 float4s, 8 per thread, no VGPR staging.
#pragma unroll
    for (int j = 0; j < 8; ++j) {
        int idx = (tid + j * 256) * 4; // float index, 16B-aligned
        unsigned int lds_off   = (unsigned int)(uintptr_t)&ldsP[idx];
        const float* gaddr     = protos + idx;
        asm volatile("global_load_async_to_lds_b128 %0, %1, off"
                     :
                     : "v"(lds_off), "v"(gaddr)
                     : "memory");
    }
    if (tid < NCLS) ldsPn[tid] = pnorm2[tid];
    asm volatile("s_wait_asynccnt 0x0" ::: "memory");
    __syncthreads();

    int lane   = tid & 31;
    int wave   = tid >> 5;
    int laneLo = lane & 15;
    int grp    = lane >> 4;  // half-wave id: selects K-pair and row offset
    int koff   = grp * 2;
    int qbase  = blockIdx.x * 128 + wave * 16;

    int qrow = qbase + laneLo;
    if (qrow >= Nq) qrow = Nq - 1;
    const float* qptr = Q + (size_t)qrow * DIM + koff;

    v8f acc[4];
#pragma unroll
    for (int t = 0; t < 4; ++t)
        acc[t] = (v8f){0.f, 0.f, 0.f, 0.f, 0.f, 0.f, 0.f, 0.f};

    float qn2 = 0.0f;

#pragma unroll 4
    for (int k = 0; k < DIM; k += 4) {
        v2f a = *(const v2f*)(qptr + k);          // A fragment (global, read once)
        qn2 += a.x * a.x + a.y * a.y;             // this lane's half of ||q||^2
#pragma unroll
        for (int t = 0; t < 4; ++t) {
            v2f b = *(const v2f*)(&ldsP[(t * 16 + laneLo) * DIM + k + koff]);
            acc[t] = __builtin_amdgcn_wmma_f32_16x16x4_f32(
                false, a, false, b, (short)0, acc[t], false, false);
        }
    }

    // complete ||q||^2: partner half-wave holds the other 64 dims
    qn2 += __shfl_xor(qn2, 16, 32);

    float pn[4];
#pragma unroll
    for (int t = 0; t < 4; ++t) pn[t] = ldsPn[t * 16 + laneLo];

#pragma unroll
    for (int r = 0; r < 8; ++r) {
        // row handled by this (component, half-wave): grp*8 + r
        float qn = __shfl(qn2, grp * 8 + r, 32); // lane m holds ||q_m||^2
        float dd[4];
        float mn = 3.0e38f;
#pragma unroll
        for (int t = 0; t < 4; ++t) {
            float sq = qn + pn[t] - 2.0f * acc[t][r];
            dd[t]    = sqrtf(fmaxf(sq, 0.0f));
            mn       = fminf(mn, dd[t]);
        }
        // row-min across the 16 lanes of this half-wave (xor < 16 stays inside)
        mn = fminf(mn, __shfl_xor(mn, 1, 32));
        mn = fminf(mn, __shfl_xor(mn, 2, 32));
        mn = fminf(mn, __shfl_xor(mn, 4, 32));
        mn = fminf(mn, __shfl_xor(mn, 8, 32));

        float e[4], s = 0.0f;
#pragma unroll
        for (int t = 0; t < 4; ++t) {
            e[t] = __expf(mn - dd[t]); // softmax(-d) = exp(min_d - d)/sum
            s += e[t];
        }
        s += __shfl_xor(s, 1, 32);
        s += __shfl_xor(s, 2, 32);
        s += __shfl_xor(s, 4, 32);
        s += __shfl_xor(s, 8, 32);
        float inv = 1.0f / s;

        int row = qbase + grp * 8 + r;
        if (row < Nq) {
#pragma unroll
            for (int t = 0; t < 4; ++t)
                out[(size_t)row * NCLS + t * 16 + laneLo] = e[t] * inv;
        }
    }
}

// ---------------------------------------------------------------------------
extern "C" void kernel_launch(void* const* d_in, const int* in_sizes, int n_in,
                              void* d_out, int out_size, void* d_ws, size_t ws_size,
                              hipStream_t stream) {
    const float* se = (const float*)d_in[0]; // support_embeddings [Ns,128]
    const int*   sl = (const int*)d_in[1];   // support_labels [Ns]
    const float* qe = (const float*)d_in[2]; // query_embeddings [Nq,128]
    float* out = (float*)d_out;
    float* ws  = (float*)d_ws;

    int Ns = in_sizes[1];
    int Nq = in_sizes[2] / DIM;

    float* sums   = ws;              // 8192 f32 -> prototypes after finalize
    float* counts = ws + NCLS * DIM; // 64 f32   -> ||p||^2 after finalize

    fsl_zero_ws<<<(WS_FLOATS + 255) / 256, 256, 0, stream>>>(ws, WS_FLOATS);
    fsl_proto_accum<<<(Ns + 255) / 256, 256, 0, stream>>>(se, sl, sums, counts, Ns);
    fsl_proto_final<<<NCLS, DIM, 0, stream>>>(sums, counts);
    fsl_query_softmax<<<(Nq + 127) / 128, 256, 0, stream>>>(qe, sums, counts, out, Nq);
}